// FCOS3DTarget_83021717831985
// MI455X (gfx1250) — compile-verified
//
#include <hip/hip_runtime.h>
#include <math.h>

// ---------------- problem constants (from reference) ----------------
#define NPTS   30929      // 116*200 + 58*100 + 29*50 + 15*25 + 8*13
#define M_BOX  64
#define BATCH  4
#define BN     (BATCH * NPTS)
#define BLK    128        // 4 wave32 waves per block
#define D2S    65         // LDS stride for d2 tile: 65 -> bank = (row+col)%64, conflict free

typedef float v2f __attribute__((ext_vector_type(2)));
typedef float v8f __attribute__((ext_vector_type(8)));

struct PtInfo { float x, y, s, rr0, rr1; int pos; };

// Decode a global (per-batch) point index into coordinates / stride /
// regress-range / output position. Level tables are compile-time constants.
__device__ __forceinline__ PtInfo decode_pt(int n, int b) {
    PtInfo p;
    int local, w, si, off4, npl;
    if (n < 23200)      { local = n;         w = 200; si = 8;   p.rr0 = -1.0f;  p.rr1 = 48.0f;  off4 = 0;         npl = 23200; }
    else if (n < 29000) { local = n - 23200; w = 100; si = 16;  p.rr0 = 48.0f;  p.rr1 = 96.0f;  off4 = 4 * 23200; npl = 5800;  }
    else if (n < 30450) { local = n - 29000; w = 50;  si = 32;  p.rr0 = 96.0f;  p.rr1 = 192.0f; off4 = 4 * 29000; npl = 1450;  }
    else if (n < 30825) { local = n - 30450; w = 25;  si = 64;  p.rr0 = 192.0f; p.rr1 = 384.0f; off4 = 4 * 30450; npl = 375;   }
    else                { local = n - 30825; w = 13;  si = 128; p.rr0 = 384.0f; p.rr1 = 1e8f;   off4 = 4 * 30825; npl = 104;   }
    int row = local / w;
    int col = local - row * w;
    p.x = (float)(col * si + si / 2);
    p.y = (float)(row * si + si / 2);
    p.s = (float)si;
    p.pos = off4 + b * npl + local;
    return p;
}

__global__ __launch_bounds__(BLK)
void fcos3d_target_kernel(const float* __restrict__ gt_bboxes,   // (B,M,4)
                          const float* __restrict__ g3d,         // (B,M,9)
                          const int*   __restrict__ gl3d,        // (B,M)
                          const float* __restrict__ c2d,         // (B,M,2)
                          const float* __restrict__ depths,      // (B,M)
                          const int*   __restrict__ attrs,       // (B,M)
                          float*       __restrict__ out)
{
    __shared__ float s_d2[BLK * D2S];                 // 128x64 d^2 tile (padded)
    __shared__ float s_cx[M_BOX], s_cy[M_BOX];
    __shared__ float s_n2cx[M_BOX], s_n2cy[M_BOX], s_c2[M_BOX];  // WMMA B-matrix terms
    __shared__ float s_x1[M_BOX], s_y1[M_BOX], s_x2[M_BOX], s_y2[M_BOX];
    __shared__ float s_dep[M_BOX];
    __shared__ float s_tail[M_BOX][6];                // g3d[3:9] with yaw fixup
    __shared__ int   s_lab[M_BOX], s_attr[M_BOX];

    const int b    = blockIdx.y;
    const int tid  = threadIdx.x;
    const int base = blockIdx.x * BLK;

    // ---------------- Stage A: stage the 64 GT boxes into LDS ----------------
    if (tid < M_BOX) {
        const int m = tid;
        const float* bb = gt_bboxes + ((size_t)b * M_BOX + m) * 4;
        const float* cc = c2d       + ((size_t)b * M_BOX + m) * 2;
        const float* gg = g3d       + ((size_t)b * M_BOX + m) * 9;
        const float cx = cc[0], cy = cc[1];
        s_cx[m] = cx; s_cy[m] = cy;
        const float xc = cx - 800.0f, yc = cy - 464.0f;   // centered for conditioning
        s_n2cx[m] = -2.0f * xc;
        s_n2cy[m] = -2.0f * yc;
        s_c2[m]   = xc * xc + yc * yc;
        s_x1[m] = bb[0]; s_y1[m] = bb[1]; s_x2[m] = bb[2]; s_y2[m] = bb[3];
        s_dep[m]  = depths[(size_t)b * M_BOX + m];
        const float g0 = gg[0], g2 = gg[2], g6 = gg[6];
        s_tail[m][0] = gg[3];
        s_tail[m][1] = gg[4];
        s_tail[m][2] = gg[5];
        s_tail[m][3] = -atan2f(g0, g2) + g6;              // yaw fixup
        s_tail[m][4] = gg[7];
        s_tail[m][5] = gg[8];
        s_lab[m]  = gl3d[(size_t)b * M_BOX + m];
        s_attr[m] = attrs[(size_t)b * M_BOX + m];
    }
    __syncthreads();

    // ---------------- Stage B: d^2 tiles via V_WMMA_F32_16X16X4_F32 ----------
    // d^2 = (x^2+y^2)*1 + x*(-2cx) + y*(-2cy) + 1*(cx^2+cy^2)   (K = 4)
    const int lane = tid & 31;
    const int wv   = tid >> 5;
    const int l15  = lane & 15;
    const int hi   = lane >> 4;      // 0: lanes 0-15, 1: lanes 16-31

#pragma unroll
    for (int t = 0; t < 2; ++t) {
        const int pt = wv * 32 + t * 16;                  // point-tile base (0..112)
        int n = base + pt + l15;
        n = (n < NPTS) ? n : (NPTS - 1);                  // clamp tail (EXEC stays full)
        const PtInfo pi = decode_pt(n, b);
        const float xc = pi.x - 800.0f, yc = pi.y - 464.0f;
        // A 16x4 f32 layout: lanes 0-15 -> (K=0,K=1), lanes 16-31 -> (K=2,K=3)
        v2f a;
        a.x = hi ? yc   : (xc * xc + yc * yc);
        a.y = hi ? 1.0f : xc;
#pragma unroll
        for (int bt = 0; bt < 4; ++bt) {
            const int bx = bt * 16;                       // box-tile base
            const int m  = bx + l15;
            // B 4x16 f32 layout mirrors A: lanes -> N, (VGPR, lane-half) -> K
            v2f bm;
            bm.x = hi ? s_n2cy[m] : 1.0f;
            bm.y = hi ? s_c2[m]   : s_n2cx[m];
            v8f c = {};
            v8f d = __builtin_amdgcn_wmma_f32_16x16x4_f32(
                        false, a, false, bm, (short)0, c, false, false);
            // D layout: VGPR r -> M = r + 8*hi ; N = lane&15
#pragma unroll
            for (int r = 0; r < 8; ++r) {
                const int row = pt + hi * 8 + r;
                s_d2[row * D2S + bx + l15] = d[r];
            }
        }
    }
    __syncthreads();

    // ---------------- Stage C: masked argmin over M=64 boxes ------------------
    const int n = base + tid;
    const PtInfo p = decode_pt((n < NPTS) ? n : (NPTS - 1), b);
    const float r = p.s * 1.5f;                           // stride * RADIUS
    float best = 1e8f;                                    // INF sentinel (d^2 <= 3.4e6 real)
    int   bi   = 0;
#pragma unroll 4
    for (int m = 0; m < M_BOX; ++m) {
        const float d2  = s_d2[tid * D2S + m];
        const float l   = p.x - s_x1[m];
        const float tp  = p.y - s_y1[m];
        const float rt  = s_x2[m] - p.x;
        const float bt  = s_y2[m] - p.y;
        const float mrd = fmaxf(fmaxf(l, tp), fmaxf(rt, bt));
        const float adx = fabsf(p.x - s_cx[m]);
        const float ady = fabsf(p.y - s_cy[m]);
        // cb_min > 0  <=>  max(|dx|,|dy|) < r   (exact rewrite)
        const bool cond = (fmaxf(adx, ady) < r) && (mrd >= p.rr0) && (mrd <= p.rr1);
        const float val = cond ? d2 : 1e8f;
        if (val < best) { best = val; bi = m; }           // strict < = argmin first-occurrence
    }

    // ---------------- Stage D: gather + outputs -------------------------------
    if (n < NPTS) {
        const bool  bg = (best >= 1e8f);
        const float dx = p.x - s_cx[bi];
        const float dy = p.y - s_cy[bi];
        const float rel  = sqrtf(dx * dx + dy * dy) / (1.414f * p.s * 1.5f);
        const float cent = expf(-2.5f * rel);             // ALPHA = 2.5

        float* out_l = out;                               // (B*N,)
        float* out_b = out + BN;                          // (B*N, 9)
        float* out_c = out + (size_t)10 * BN;             // (B*N,)
        float* out_a = out + (size_t)11 * BN;             // (B*N,)

        out_l[p.pos] = bg ? 10.0f : (float)s_lab[bi];     // NUM_CLASSES
        out_c[p.pos] = cent;
        out_a[p.pos] = bg ? 9.0f  : (float)s_attr[bi];    // NUM_ATTRS

        float* bo = out_b + (size_t)p.pos * 9;
        bo[0] = dx / p.s;                                 // stride is power of 2: exact
        bo[1] = dy / p.s;
        bo[2] = s_dep[bi];
        bo[3] = s_tail[bi][0];
        bo[4] = s_tail[bi][1];
        bo[5] = s_tail[bi][2];
        bo[6] = s_tail[bi][3];
        bo[7] = s_tail[bi][4];
        bo[8] = s_tail[bi][5];
    }
}

extern "C" void kernel_launch(void* const* d_in, const int* in_sizes, int n_in,
                              void* d_out, int out_size, void* d_ws, size_t ws_size,
                              hipStream_t stream) {
    // setup_inputs() order:
    // 0: gt_bboxes (B,M,4) f32   1: gt_labels (B,M) i32 (unused by reference outputs)
    // 2: gt_bboxes_3d (B,M,9) f32  3: gt_labels_3d (B,M) i32
    // 4: centers2d (B,M,2) f32   5: depths (B,M) f32   6: attr_labels (B,M) i32
    const float* gt_bboxes = (const float*)d_in[0];
    const float* g3d       = (const float*)d_in[2];
    const int*   gl3d      = (const int*)d_in[3];
    const float* c2d       = (const float*)d_in[4];
    const float* depths    = (const float*)d_in[5];
    const int*   attrs     = (const int*)d_in[6];

    dim3 grid((NPTS + BLK - 1) / BLK, BATCH);  // (242, 4)
    fcos3d_target_kernel<<<grid, dim3(BLK), 0, stream>>>(
        gt_bboxes, g3d, gl3d, c2d, depths, attrs, (float*)d_out);
}